// ksparse_69432441307894
// MI455X (gfx1250) — compile-verified
//
#include <hip/hip_runtime.h>

// k-sparse row threshold: y = x * (x >= theta_row), theta = 2867th-smallest of 4096.
// One workgroup (256 threads, 8 wave32) per row. Async global->LDS staging
// (ASYNCcnt DMA), register-resident 4-pass radix select; bucket selection done
// by a single wave32 shuffle-scan (3 barriers/pass instead of 18).

#define BATCH 16384
#define UNITS 4096
#define KIDX  2867          // int(0.7 * 4096)
#define TPB   256

#if defined(__gfx1250__) && __has_builtin(__builtin_amdgcn_global_load_async_to_lds_b128)
#define USE_ASYNC_LDS 1
#else
#define USE_ASYNC_LDS 0
#endif

// 16-byte int vector, as expected by the async-to-LDS builtin
typedef __attribute__((__vector_size__(16))) int vi4;
typedef __attribute__((address_space(1))) vi4 gvi4;   // global
typedef __attribute__((address_space(3))) vi4 svi4;   // LDS

__device__ __forceinline__ unsigned f2key(float x) {
  // monotone float -> u32 map: negatives -> [0, 0x7FFFFFFF], positives -> [0x80000000, ...]
  unsigned u = __float_as_uint(x);
  return (u & 0x80000000u) ? ~u : (u | 0x80000000u);
}

__global__ __launch_bounds__(TPB) void ksparse_rowselect_kernel(
    const float* __restrict__ X, float* __restrict__ Y) {
  __shared__ __align__(16) float    rowv[UNITS];  // 16 KB staging for one row
  __shared__ __align__(16) unsigned hist[256];    // radix histogram
  __shared__ unsigned sel[2];                     // [0]=digit, [1]=rank within bucket

  const int    tid     = (int)threadIdx.x;
  const size_t rowbase = (size_t)blockIdx.x * (size_t)UNITS;
  const float* src     = X + rowbase;
  float*       dst     = Y + rowbase;

  // ---- Stage row into LDS (async DMA path on CDNA5) --------------------
#if USE_ASYNC_LDS
  #pragma unroll
  for (int i = 0; i < 4; ++i) {
    const int e = (i * TPB + tid) * 4;  // element index of this 16B chunk
    __builtin_amdgcn_global_load_async_to_lds_b128(
        (gvi4*)(const_cast<float*>(src) + e),
        (svi4*)(&rowv[e]),
        0, 0);
  }
  #if __has_builtin(__builtin_amdgcn_s_wait_asynccnt)
  __builtin_amdgcn_s_wait_asynccnt(0);
  #else
  asm volatile("s_wait_asynccnt 0" ::: "memory");
  #endif
#else
  #pragma unroll
  for (int i = 0; i < 4; ++i) {
    const int c = i * TPB + tid;
    ((float4*)rowv)[c] = ((const float4*)src)[c];
  }
#endif
  __syncthreads();

  // ---- Pull this thread's 16 elements into registers -------------------
  float4 f[4];
  #pragma unroll
  for (int i = 0; i < 4; ++i) f[i] = ((float4*)rowv)[i * TPB + tid];

  unsigned keys[16];
  #pragma unroll
  for (int i = 0; i < 4; ++i) {
    keys[4 * i + 0] = f2key(f[i].x);
    keys[4 * i + 1] = f2key(f[i].y);
    keys[4 * i + 2] = f2key(f[i].z);
    keys[4 * i + 3] = f2key(f[i].w);
  }

  // ---- 4-pass MSB-first radix select of the KIDX-th smallest key -------
  unsigned prefix = 0u, msk = 0u, k = KIDX;
  #pragma unroll 1
  for (int pass = 0; pass < 4; ++pass) {
    const int shift = 24 - 8 * pass;

    hist[tid] = 0u;
    __syncthreads();

    #pragma unroll
    for (int j = 0; j < 16; ++j) {
      const unsigned key = keys[j];
      if ((key & msk) == prefix) atomicAdd(&hist[(key >> shift) & 0xFFu], 1u);
    }
    __syncthreads();

    // Wave 0 alone selects the bucket containing rank k: each lane owns 8 bins,
    // wave32 shuffle scan gives the running base -- no extra workgroup barriers.
    if (tid < 32) {
      const int lane = tid;
      const uint4 h0 = ((const uint4*)hist)[lane * 2 + 0];  // bins 8l .. 8l+3
      const uint4 h1 = ((const uint4*)hist)[lane * 2 + 1];  // bins 8l+4 .. 8l+7
      unsigned c[8] = {h0.x, h0.y, h0.z, h0.w, h1.x, h1.y, h1.z, h1.w};
      unsigned tot = 0u;
      #pragma unroll
      for (int j = 0; j < 8; ++j) tot += c[j];

      unsigned incl = tot;                    // inclusive scan across 32 lanes
      #pragma unroll
      for (int d = 1; d < 32; d <<= 1) {
        const unsigned t = __shfl_up(incl, d, 32);
        if (lane >= d) incl += t;
      }
      unsigned excl = incl - tot;             // rank before this lane's 8 bins
      #pragma unroll
      for (int j = 0; j < 8; ++j) {
        const unsigned nxt = excl + c[j];
        if (k >= excl && k < nxt) {           // exactly one (lane,bin) hits
          sel[0] = (unsigned)(lane * 8 + j);
          sel[1] = k - excl;
        }
        excl = nxt;
      }
    }
    __syncthreads();
    prefix |= sel[0] << shift;
    msk    |= 0xFFu << shift;
    k       = sel[1];
  }

  // ---- Invert key -> theta, apply threshold, coalesced stores ----------
  const unsigned tk = prefix;
  const unsigned ub = (tk & 0x80000000u) ? (tk & 0x7FFFFFFFu) : ~tk;
  const float theta = __uint_as_float(ub);

  #pragma unroll
  for (int i = 0; i < 4; ++i) {
    float4 v = f[i];
    v.x = (v.x >= theta) ? v.x : 0.0f;
    v.y = (v.y >= theta) ? v.y : 0.0f;
    v.z = (v.z >= theta) ? v.z : 0.0f;
    v.w = (v.w >= theta) ? v.w : 0.0f;
    ((float4*)dst)[i * TPB + tid] = v;  // lanes contiguous -> 128-bit coalesced
  }
}

extern "C" void kernel_launch(void* const* d_in, const int* in_sizes, int n_in,
                              void* d_out, int out_size, void* d_ws, size_t ws_size,
                              hipStream_t stream) {
  (void)in_sizes; (void)n_in; (void)d_ws; (void)ws_size; (void)out_size;
  const float* X = (const float*)d_in[0];
  float*       Y = (float*)d_out;
  hipLaunchKernelGGL(ksparse_rowselect_kernel, dim3(BATCH), dim3(TPB), 0, stream, X, Y);
}